// RepeatHeads_33011118637634
// MI455X (gfx1250) — compile-verified
//
#include <hip/hip_runtime.h>
#include <hip/hip_bf16.h>

typedef __attribute__((ext_vector_type(2))) float v2f;
typedef __attribute__((ext_vector_type(8))) float v8f;

typedef __attribute__((address_space(1))) int g_i32;   // global (AS1)
typedef __attribute__((address_space(3))) int l_i32;   // LDS (AS3)

#define B_ 8
#define E_ 512
#define S_ 1024
#define H_ 8

// ---------------------------------------------------------------------------
// Mt[h][k][j] = sum_{d >= k} mix_w[h,d] * out_w[j,d]   (K-major for TN GEMM)
// ---------------------------------------------------------------------------
__global__ void __launch_bounds__(256) build_mt(const float* __restrict__ mix_w,
                                                const float* __restrict__ out_w,
                                                float* __restrict__ Mt) {
  const int t = blockIdx.x * 256 + threadIdx.x;   // 0 .. H*S-1
  const int h = t >> 10;
  const int j = t & (S_ - 1);
  const float* mw = mix_w + h * S_;
  const float* ow = out_w + (size_t)j * S_;
  float* mt = Mt + (size_t)h * S_ * S_ + j;
  float acc = 0.f;
  for (int k = S_ - 1; k >= 0; --k) {
    acc += mw[k] * ow[k];
    mt[(size_t)k * S_] = acc;           // coalesced across j
  }
}

// ---------------------------------------------------------------------------
// c[h][j] = out_b[j] + sum_d proj_b[h,d]*M_h[j,d] + sum_d mix_b[h,d]*out_w[j,d]
// ---------------------------------------------------------------------------
__global__ void __launch_bounds__(256) build_c(const float* __restrict__ proj_b,
                                               const float* __restrict__ mix_b,
                                               const float* __restrict__ out_w,
                                               const float* __restrict__ out_b,
                                               const float* __restrict__ Mt,
                                               float* __restrict__ cvec) {
  const int t = blockIdx.x * 256 + threadIdx.x;   // 0 .. H*S-1
  const int h = t >> 10;
  const int j = t & (S_ - 1);
  const float* pb = proj_b + h * S_;
  const float* mb = mix_b + h * S_;
  const float* ow = out_w + (size_t)j * S_;
  const float* mt = Mt + (size_t)h * S_ * S_ + j;
  float acc = out_b[j];
  for (int d = 0; d < S_; ++d)
    acc += pb[d] * mt[(size_t)d * S_] + mb[d] * ow[d];
  cvec[h * S_ + j] = acc;
}

// ---------------------------------------------------------------------------
// Xt[b][s][e] = X[b][e][s] — 32x33 LDS-tiled transpose; fill uses the gfx1250
// async global->LDS path (GLOBAL_LOAD_ASYNC_TO_LDS_B32, ASYNCcnt) when the
// builtin is available.
// ---------------------------------------------------------------------------
__global__ void __launch_bounds__(256) transpose_x(const float* __restrict__ X,
                                                   float* __restrict__ Xt) {
  __shared__ float tile[32][33];
  const int b  = blockIdx.z;
  const int s0 = blockIdx.x * 32;
  const int e0 = blockIdx.y * 32;
  const int tx = threadIdx.x;          // 0..31 (contiguous in s on load)
  const int ty = threadIdx.y;          // 0..7
  const float* src = X + (size_t)b * E_ * S_;
  float* dst = Xt + (size_t)b * E_ * S_;
#pragma unroll
  for (int i = 0; i < 32; i += 8) {
    const float* gp = src + (size_t)(e0 + ty + i) * S_ + s0 + tx;
#if __has_builtin(__builtin_amdgcn_global_load_async_to_lds_b32)
    __builtin_amdgcn_global_load_async_to_lds_b32(
        (g_i32*)gp, (l_i32*)&tile[ty + i][tx], 0, 0);
#else
    tile[ty + i][tx] = *gp;
#endif
  }
#if __has_builtin(__builtin_amdgcn_global_load_async_to_lds_b32)
#if __has_builtin(__builtin_amdgcn_s_wait_asynccnt)
  __builtin_amdgcn_s_wait_asynccnt(0);
#else
  asm volatile("s_wait_asynccnt 0x0" ::: "memory");
#endif
#endif
  __syncthreads();
#pragma unroll
  for (int i = 0; i < 32; i += 8) {
    dst[(size_t)(s0 + ty + i) * E_ + e0 + tx] = tile[tx][ty + i];
  }
}

// ---------------------------------------------------------------------------
// GEMM1 (TN): Qt[h][s][j] = sum_d Wp[h][d][s] * Mt[h][d][j]
// A = proj_w[h] layout [K=d][M=s], ld=S ; B = Mt[h] layout [K=d][N=j], ld=S
// One wave computes a 16x64 strip via V_WMMA_F32_16X16X4_F32; all global
// traffic is 64B-coalesced per half-wave.
// ---------------------------------------------------------------------------
__global__ void __launch_bounds__(256) gemm_tn_qt(const float* __restrict__ Wp,
                                                  const float* __restrict__ Mt,
                                                  float* __restrict__ Qt) {
  const int wave = (blockIdx.x * 256 + threadIdx.x) >> 5;
  const int lane = threadIdx.x & 31;
  const int l16  = lane & 15;
  const int kh   = (lane >> 4) << 1;            // K half per half-wave: 0 or 2
  const int h    = wave >> 10;                  // 1024 wave-tasks per head
  const int rem  = wave & 1023;
  const int m0   = (rem >> 4) << 4;             // 64 s-tiles of 16
  const int n0   = (rem & 15) << 6;             // 16 j-groups of 64
  const float* A  = Wp + (size_t)h * S_ * S_;
  const float* Bp = Mt + (size_t)h * S_ * S_;
  v8f acc0 = {}, acc1 = {}, acc2 = {}, acc3 = {};
#pragma unroll 4
  for (int k0 = 0; k0 < S_; k0 += 4) {
    const float* Ak = A  + (size_t)(k0 + kh) * S_;
    const float* Bk = Bp + (size_t)(k0 + kh) * S_;
    __builtin_prefetch(Ak + 8 * S_ + m0, 0, 1);
    v2f a;  a.x  = Ak[m0 + l16];            a.y  = Ak[S_ + m0 + l16];
    v2f b0; b0.x = Bk[n0 + l16];            b0.y = Bk[S_ + n0 + l16];
    v2f b1; b1.x = Bk[n0 + 16 + l16];       b1.y = Bk[S_ + n0 + 16 + l16];
    v2f b2; b2.x = Bk[n0 + 32 + l16];       b2.y = Bk[S_ + n0 + 32 + l16];
    v2f b3; b3.x = Bk[n0 + 48 + l16];       b3.y = Bk[S_ + n0 + 48 + l16];
    acc0 = __builtin_amdgcn_wmma_f32_16x16x4_f32(false, a, false, b0, (short)0, acc0, false, false);
    acc1 = __builtin_amdgcn_wmma_f32_16x16x4_f32(false, a, false, b1, (short)0, acc1, false, false);
    acc2 = __builtin_amdgcn_wmma_f32_16x16x4_f32(false, a, false, b2, (short)0, acc2, false, false);
    acc3 = __builtin_amdgcn_wmma_f32_16x16x4_f32(false, a, false, b3, (short)0, acc3, false, false);
  }
  float* C = Qt + (size_t)h * S_ * S_;
  const int rbase = m0 + ((lane >> 4) << 3);    // lanes 16-31 hold rows M+8..M+15
#pragma unroll
  for (int r = 0; r < 8; ++r) {
    const size_t row = (size_t)(rbase + r) * S_;
    C[row + n0      + l16] = acc0[r];
    C[row + n0 + 16 + l16] = acc1[r];
    C[row + n0 + 32 + l16] = acc2[r];
    C[row + n0 + 48 + l16] = acc3[r];
  }
}

// ---------------------------------------------------------------------------
// GEMM2 (TN): out[b, h*E+e, j] = sum_s Xt[b][s][e] * Qt[h][s][j] + c[h,j]
// A = Xt[b] layout [K=s][M=e], ld=E ; B = Qt[h] layout [K=s][N=j], ld=S
// ---------------------------------------------------------------------------
__global__ void __launch_bounds__(256) gemm_tn_out(const float* __restrict__ Xt,
                                                   const float* __restrict__ Qt,
                                                   const float* __restrict__ cvec,
                                                   float* __restrict__ out) {
  const int wave = (blockIdx.x * 256 + threadIdx.x) >> 5;
  const int lane = threadIdx.x & 31;
  const int l16  = lane & 15;
  const int kh   = (lane >> 4) << 1;
  const int bh   = wave >> 9;                   // 512 wave-tasks per (b,h)
  const int b    = bh >> 3;
  const int h    = bh & 7;
  const int rem  = wave & 511;
  const int m0   = (rem >> 4) << 4;             // 32 e-tiles of 16
  const int n0   = (rem & 15) << 6;             // 16 j-groups of 64
  const float* A  = Xt + (size_t)b * E_ * S_;   // [s][e], ld = E
  const float* Bp = Qt + (size_t)h * S_ * S_;   // [s][j], ld = S
  v8f acc0 = {}, acc1 = {}, acc2 = {}, acc3 = {};
#pragma unroll 4
  for (int k0 = 0; k0 < S_; k0 += 4) {
    const float* Ak = A  + (size_t)(k0 + kh) * E_;
    const float* Bk = Bp + (size_t)(k0 + kh) * S_;
    __builtin_prefetch(Bk + 8 * S_ + n0, 0, 1);
    v2f a;  a.x  = Ak[m0 + l16];            a.y  = Ak[E_ + m0 + l16];
    v2f b0; b0.x = Bk[n0 + l16];            b0.y = Bk[S_ + n0 + l16];
    v2f b1; b1.x = Bk[n0 + 16 + l16];       b1.y = Bk[S_ + n0 + 16 + l16];
    v2f b2; b2.x = Bk[n0 + 32 + l16];       b2.y = Bk[S_ + n0 + 32 + l16];
    v2f b3; b3.x = Bk[n0 + 48 + l16];       b3.y = Bk[S_ + n0 + 48 + l16];
    acc0 = __builtin_amdgcn_wmma_f32_16x16x4_f32(false, a, false, b0, (short)0, acc0, false, false);
    acc1 = __builtin_amdgcn_wmma_f32_16x16x4_f32(false, a, false, b1, (short)0, acc1, false, false);
    acc2 = __builtin_amdgcn_wmma_f32_16x16x4_f32(false, a, false, b2, (short)0, acc2, false, false);
    acc3 = __builtin_amdgcn_wmma_f32_16x16x4_f32(false, a, false, b3, (short)0, acc3, false, false);
  }
  const float* ch = cvec + h * S_;
  float* Cbase = out + ((size_t)b * (H_ * E_) + (size_t)h * E_ + m0) * S_;
  const int roff = (lane >> 4) << 3;
#pragma unroll
  for (int r = 0; r < 8; ++r) {
    const size_t row = (size_t)(roff + r) * S_;
    Cbase[row + n0      + l16] = acc0[r] + ch[n0      + l16];
    Cbase[row + n0 + 16 + l16] = acc1[r] + ch[n0 + 16 + l16];
    Cbase[row + n0 + 32 + l16] = acc2[r] + ch[n0 + 32 + l16];
    Cbase[row + n0 + 48 + l16] = acc3[r] + ch[n0 + 48 + l16];
  }
}

// ---------------------------------------------------------------------------
extern "C" void kernel_launch(void* const* d_in, const int* in_sizes, int n_in,
                              void* d_out, int out_size, void* d_ws, size_t ws_size,
                              hipStream_t stream) {
  const float* x      = (const float*)d_in[0];  // [B,E,S]
  const float* proj_w = (const float*)d_in[1];  // [H,HID,DIM]  ([h][d][s], d-major)
  const float* proj_b = (const float*)d_in[2];  // [H,HID]
  const float* mix_w  = (const float*)d_in[3];  // [H,S]
  const float* mix_b  = (const float*)d_in[4];  // [H,S]
  const float* out_w  = (const float*)d_in[5];  // [DIM,DIM]
  const float* out_b  = (const float*)d_in[6];  // [DIM]
  float* out = (float*)d_out;                   // [B, H*E, S]

  float* Mt = (float*)d_ws;                     // H*S*S floats (32 MB)  [h][d][j]
  float* Qt = Mt + (size_t)H_ * S_ * S_;        // H*S*S floats (32 MB)  [h][s][j]
  float* Xt = Qt + (size_t)H_ * S_ * S_;        // B*S*E floats (16 MB)  [b][s][e]
  float* cv = Xt + (size_t)B_ * S_ * E_;        // H*S  floats

  build_mt<<<(H_ * S_) / 256, 256, 0, stream>>>(mix_w, out_w, Mt);
  transpose_x<<<dim3(S_ / 32, E_ / 32, B_), dim3(32, 8), 0, stream>>>(x, Xt);
  build_c<<<(H_ * S_) / 256, 256, 0, stream>>>(proj_b, mix_b, out_w, out_b, Mt, cv);
  // Qt_h = Wp_h^T x M_h^T : 8 * (64 s-tiles * 16 j-groups) = 8192 waves
  gemm_tn_qt<<<(H_ * (S_ / 16) * (S_ / 64)) / 8, 256, 0, stream>>>(proj_w, Mt, Qt);
  // out = Xt^T @ Qt + c : 64 * (32 * 16) = 32768 waves
  gemm_tn_out<<<(B_ * H_ * (E_ / 16) * (S_ / 64)) / 8, 256, 0, stream>>>(Xt, Qt, cv, out);
}